// DownSampler_12163347382448
// MI455X (gfx1250) — compile-verified
//
#include <hip/hip_runtime.h>
#include <hip/hip_bf16.h>

typedef float v2f __attribute__((ext_vector_type(2)));
typedef float v8f __attribute__((ext_vector_type(8)));

#define H20   1032
#define OUT20 516
#define OUT60 172
#define KS    41
#define PAD   20

__device__ __forceinline__ int refl1032(int t) {
  t = (t < 0) ? -t : t;
  return (t >= 1032) ? (2062 - t) : t;   // 2*(1032-1) - t
}

// ---------------------------------------------------------------------------
// Path 20: depthwise 41x41 conv fused with nearest-exact /2 decimation,
// computed only at sampled points  out(Y,X) = sum_ky,kx xp(2Y+1+ky, 2X+1+kx)*w(ky,kx)
// via V_WMMA_F32_16X16X4_F32:
//   D[m,n] += sum_j A[m,j]*B[j,n]      (accumulated over ky, j in [0,72))
//   A[m,j] = xp(2(Y0+m)+1+ky, 2X0+1+j)          (input rows, from LDS)
//   B[j,n] = w(ky, j-2n) if 0<=j-2n<41 else 0   (banded Toeplitz, zero-padded LDS)
// One 128-thread block = 4 waves = 32x32 output tile sharing one LDS input tile.
// ---------------------------------------------------------------------------
#define IN_ROWS  103   // 32 + 2*15 + 41
#define IN_COLS  104   // 32 + 72
#define IN_PITCH 105   // odd pitch: stride-2-row A reads hit distinct banks
#define W_PITCH  104   // 32 zeros | 41 taps | 31 zeros
#define W_OFF    32

__global__ __launch_bounds__(128) void conv20_wmma_kernel(
    const float* __restrict__ x, const float* __restrict__ w, float* __restrict__ out)
{
  __shared__ float sIn[IN_ROWS * IN_PITCH];   // 43260 B
  __shared__ float sW[KS * W_PITCH];          // 17056 B  (total 60316 < 64KB)

  const int tilesX = 17;                      // ceil(516/32)
  const int tX = blockIdx.x % tilesX;
  const int tY = blockIdx.x / tilesX;
  const int bc = blockIdx.y;                  // b*6 + c
  const float* xin = x + (size_t)bc * (H20 * H20);
  const float* wch = w + (size_t)(bc % 6) * (KS * KS);

  const int tid = threadIdx.x;                // 0..127

  // Cooperative reflect-padded input gather into LDS.
  const int rowBase = 64 * tY + 1;            // padded-coord of sIn row 0
  const int colBase = 64 * tX + 1;
  for (int i = tid; i < IN_ROWS * IN_COLS; i += 128) {
    int r = i / IN_COLS, c = i - r * IN_COLS;
    int sr = refl1032(rowBase + r - PAD);
    int sc = refl1032(colBase + c - PAD);
    sIn[r * IN_PITCH + c] = xin[(size_t)sr * H20 + sc];
  }
  // Zero-padded banded weight rows.
  for (int i = tid; i < KS * W_PITCH; i += 128) sW[i] = 0.0f;
  __syncthreads();
  for (int i = tid; i < KS * KS; i += 128) {
    int ky = i / KS, kx = i - ky * KS;
    sW[ky * W_PITCH + W_OFF + kx] = wch[i];
  }
  __syncthreads();

  const int wave = tid >> 5;                  // quadrant of the 32x32 tile
  const int lane = tid & 31;
  const int qy = wave >> 1, qx = wave & 1;
  const int mn   = lane & 15;                 // A row M == B/D column N
  const int koff = (lane >> 4) << 1;          // lanes16-31 hold K+2,K+3 (ISA A/B layout)

  v8f vz = {0.f,0.f,0.f,0.f,0.f,0.f,0.f,0.f};
  v8f acc[4] = {vz, vz, vz, vz};              // 4 chains to hide WMMA latency

  const float* aBase = &sIn[(32 * qy + 2 * mn) * IN_PITCH + 32 * qx + koff];
  const float* bBase = &sW[W_OFF + koff - 2 * mn];

  for (int ky = 0; ky < KS; ++ky) {
    const float* ar = aBase + ky * IN_PITCH;
    const float* br = bBase + ky * W_PITCH;
#pragma unroll
    for (int jc = 0; jc < 18; ++jc) {         // K = 72 in steps of 4
      v2f a, b;
      a.x = ar[4 * jc];  a.y = ar[4 * jc + 1];   // -> ds_read2_b32
      b.x = br[4 * jc];  b.y = br[4 * jc + 1];
      acc[jc & 3] = __builtin_amdgcn_wmma_f32_16x16x4_f32(
          false, a, false, b, (short)0, acc[jc & 3], false, false);
    }
  }

  v8f s = acc[0] + acc[1] + acc[2] + acc[3];

  // D layout: VGPR v holds rows v (lanes 0-15) and v+8 (lanes 16-31), col = lane&15.
  const int X = 32 * tX + 16 * qx + mn;
  const int Ybase = 32 * tY + 16 * qy + ((lane >> 4) << 3);
  if (X < OUT20) {
    float* op = out + (size_t)bc * (OUT20 * OUT20);
#pragma unroll
    for (int v = 0; v < 8; ++v) {
      int Y = Ybase + v;
      if (Y < OUT20) op[(size_t)Y * OUT20 + X] = s[v];
    }
  }
}

// ---------------------------------------------------------------------------
// Path 60: stride-6 decimated depthwise conv (0.8 GFLOP) — LDS-tiled direct FMA.
// out(Y,X) = sum xp(6Y+3+ky, 6X+3+kx) * w(ky,kx)
// ---------------------------------------------------------------------------
#define IN60_ROWS  83    // 6*7 + 41
#define IN60_COLS  131   // 6*15 + 41
#define IN60_PITCH 132

__global__ __launch_bounds__(128) void conv60_kernel(
    const float* __restrict__ x, const float* __restrict__ w, float* __restrict__ out)
{
  __shared__ float sIn[IN60_ROWS * IN60_PITCH];  // 43824 B
  __shared__ float sW[KS * KS];                  // 6724 B

  const int X0 = blockIdx.x * 16;
  const int Y0 = blockIdx.y * 8;
  const int bc = blockIdx.z;                     // b*2 + c
  const float* xin = x + (size_t)bc * (H20 * H20);
  const float* wch = w + (size_t)(bc % 2) * (KS * KS);

  const int tx = threadIdx.x, ty = threadIdx.y;
  const int tid = ty * 16 + tx;

  const int rowBase = 6 * Y0 + 3;
  const int colBase = 6 * X0 + 3;
  for (int i = tid; i < IN60_ROWS * IN60_COLS; i += 128) {
    int r = i / IN60_COLS, c = i - r * IN60_COLS;
    sIn[r * IN60_PITCH + c] =
        xin[(size_t)refl1032(rowBase + r - PAD) * H20 + refl1032(colBase + c - PAD)];
  }
  for (int i = tid; i < KS * KS; i += 128) sW[i] = wch[i];
  __syncthreads();

  float acc = 0.0f;
  for (int ky = 0; ky < KS; ++ky) {
    const float* rp = &sIn[(6 * ty + ky) * IN60_PITCH + 6 * tx];
    const float* wp = &sW[ky * KS];
#pragma unroll
    for (int kx = 0; kx < KS; ++kx) acc = fmaf(rp[kx], wp[kx], acc);
  }

  const int X = X0 + tx, Y = Y0 + ty;
  if (X < OUT60 && Y < OUT60)
    out[(size_t)bc * (OUT60 * OUT60) + (size_t)Y * OUT60 + X] = acc;
}

// ---------------------------------------------------------------------------
extern "C" void kernel_launch(void* const* d_in, const int* in_sizes, int n_in,
                              void* d_out, int out_size, void* d_ws, size_t ws_size,
                              hipStream_t stream) {
  (void)in_sizes; (void)n_in; (void)out_size; (void)d_ws; (void)ws_size;

  const float* x20 = (const float*)d_in[0];   // [4,6,1032,1032]
  const float* x60 = (const float*)d_in[1];   // [4,2,1032,1032]
  const float* w20 = (const float*)d_in[2];   // [6,1,41,41]
  const float* w60 = (const float*)d_in[3];   // [2,1,41,41]

  float* out20 = (float*)d_out;                             // [4,6,516,516]
  float* out60 = out20 + (size_t)4 * 6 * OUT20 * OUT20;     // [4,2,172,172]

  dim3 g20(17 * 17, 24);          // 32x32 output tiles x (b,c)
  conv20_wmma_kernel<<<g20, 128, 0, stream>>>(x20, w20, out20);

  dim3 g60(11, 22, 8);            // 16x8 output tiles x (b,c)
  conv60_kernel<<<g60, dim3(16, 8), 0, stream>>>(x60, w60, out60);
}